// minus_rate_function_8246337209007
// MI455X (gfx1250) — compile-verified
//
#include <hip/hip_runtime.h>
#include <math.h>

typedef __attribute__((ext_vector_type(2))) float v2f;
typedef __attribute__((ext_vector_type(8))) float v8f;

// Problem dims
#define KUSR 128
#define MDIM 1024
#define NDIM 128
#define NROWS (KUSR * MDIM)     // 131072 flat (k,m) rows
#define N0F   1e-11f            // 10^(-80/10)/1000

// ws layout (floats): h_re | h_im | g_re | g_im
#define WS_HRE 0
#define WS_HIM 131072
#define WS_GRE 262144
#define WS_GIM (262144 + 16384)

// ---------------------------------------------------------------------------
// Kernel 1: h = h_d + A·v  (fused complex GEMV over the 134 MB A stream)
// One wave per 16 flat rows; V_WMMA_F32_16X16X4_F32 with v_re/v_im in B cols 0/1.
// ---------------------------------------------------------------------------
__global__ __launch_bounds__(256) void k1_hvec(
    const float* __restrict__ Wv,    // (1025,256); row 0 = [v_re | v_im]
    const float* __restrict__ Are,   // (128,1024,128)
    const float* __restrict__ Aim,
    const float* __restrict__ hdre,  // (128,1024)
    const float* __restrict__ hdim,
    float* __restrict__ hre,
    float* __restrict__ him)
{
    __shared__ v2f   btab[32 * 32];      // [k-step][lane] B-operand pairs, 8 KB
    __shared__ float comb[8][4][16];     // per-wave column-combine scratch, 2 KB

    const int t    = threadIdx.x;
    const int lane = t & 31;
    const int wave = t >> 5;

    // Build B-operand table: B[k,0]=v_re[n0+k], B[k,1]=v_im[n0+k], cols 2..15 = 0.
    // Lane layout (4x16, 2 VGPRs): col = lane&15, K = vgpr + 2*(lane>=16).
    for (int e = t; e < 1024; e += 256) {
        const int i = e >> 5, l = e & 31;
        const int c = l & 15, kh = l >> 4;
        v2f b; b.x = 0.f; b.y = 0.f;
        if (c < 2) {
            const float* vp = Wv + c * 128 + 4 * i + 2 * kh;  // c==0: v_re, c==1: v_im
            b.x = vp[0];
            b.y = vp[1];
        }
        btab[e] = b;
    }
    __syncthreads();

    const int  g    = blockIdx.x * 8 + wave;            // row-group 0..8191
    const int  mrow = lane & 15;
    const long base = (long)(g * 16 + mrow) * NDIM + ((lane >> 4) << 1);
    const float* pre = Are + base;
    const float* pim = Aim + base;

    v8f acc_re = {};   // cols: [A_re·v_re, A_re·v_im, ...]
    v8f acc_im = {};   // cols: [A_im·v_re, A_im·v_im, ...]

#pragma unroll 8
    for (int i = 0; i < 32; ++i) {
        v2f a_re = *(const v2f*)(pre + 4 * i);
        v2f a_im = *(const v2f*)(pim + 4 * i);
        v2f b    = btab[i * 32 + lane];
        acc_re = __builtin_amdgcn_wmma_f32_16x16x4_f32(
            false, a_re, false, b, (short)0, acc_re, false, false);
        acc_im = __builtin_amdgcn_wmma_f32_16x16x4_f32(
            false, a_im, false, b, (short)0, acc_im, false, false);
    }

    // Columns 0/1 of D live in lanes {0,16}/{1,17}. Stage them through LDS.
    if ((lane & 15) < 2) {
        const int slot = ((lane >> 4) << 1) | (lane & 1);  // 0:l0 1:l1 2:l16 3:l17
#pragma unroll
        for (int j = 0; j < 8; ++j) {
            comb[wave][slot][j]     = acc_re[j];
            comb[wave][slot][8 + j] = acc_im[j];
        }
    }
    __syncthreads();

    if (lane < 16) {
        const int m  = lane;
        const int hi = (m >> 3) << 1;   // 0 for m<8 (lanes 0/1), 2 for m>=8 (16/17)
        const int j  = m & 7;
        const float are_vre = comb[wave][hi + 0][j];
        const float aim_vre = comb[wave][hi + 0][8 + j];
        const float are_vim = comb[wave][hi + 1][j];
        const float aim_vim = comb[wave][hi + 1][8 + j];
        const int r = g * 16 + m;
        hre[r] = hdre[r] + (are_vre - aim_vim);
        him[r] = hdim[r] + (aim_vre + are_vim);
    }
}

// ---------------------------------------------------------------------------
// Kernel 2: g = h · W (complex 128x1024x128 GEMM), one 16x16 tile per wave.
// ---------------------------------------------------------------------------
__global__ __launch_bounds__(128) void k2_gemm(
    const float* __restrict__ Wv,
    const float* __restrict__ hre,
    const float* __restrict__ him,
    float* __restrict__ gre,
    float* __restrict__ gim)
{
    const int lane = threadIdx.x & 31;
    const int wave = threadIdx.x >> 5;
    const int tile = blockIdx.x * 4 + wave;   // 0..63
    const int ti = tile >> 3, tj = tile & 7;

    const int c  = lane & 15;
    const int kh = lane >> 4;
    const long hbase = (long)(ti * 16 + c) * MDIM + 2 * kh;   // A-operand rows
    const float* wb  = Wv + 256 + (long)tj * 16 + c;          // W[r][tj*16+c] = wb[r*256]

    v8f aRR = {}, aRI = {}, aIR = {}, aII = {};

#pragma unroll 4
    for (int kc = 0; kc < MDIM; kc += 4) {
        v2f ah_re = *(const v2f*)(hre + hbase + kc);
        v2f ah_im = *(const v2f*)(him + hbase + kc);
        const long r0 = kc + 2 * kh;
        v2f b_re, b_im;
        b_re.x = wb[r0 * 256];
        b_re.y = wb[(r0 + 1) * 256];
        b_im.x = wb[r0 * 256 + 128];
        b_im.y = wb[(r0 + 1) * 256 + 128];
        aRR = __builtin_amdgcn_wmma_f32_16x16x4_f32(false, ah_re, false, b_re, (short)0, aRR, false, false);
        aRI = __builtin_amdgcn_wmma_f32_16x16x4_f32(false, ah_re, false, b_im, (short)0, aRI, false, false);
        aIR = __builtin_amdgcn_wmma_f32_16x16x4_f32(false, ah_im, false, b_re, (short)0, aIR, false, false);
        aII = __builtin_amdgcn_wmma_f32_16x16x4_f32(false, ah_im, false, b_im, (short)0, aII, false, false);
    }

    // D layout: VGPR j holds M=j (lanes 0-15) / M=j+8 (lanes 16-31), N=lane&15.
#pragma unroll
    for (int j = 0; j < 8; ++j) {
        const int  m = j + (kh << 3);
        const long o = (long)(ti * 16 + m) * 128 + tj * 16 + c;
        gre[o] = aRR[j] - aII[j];
        gim[o] = aRI[j] + aIR[j];
    }
}

// ---------------------------------------------------------------------------
// Kernel 3: SINR reduction -> scalar
// ---------------------------------------------------------------------------
__global__ __launch_bounds__(128) void k3_rate(
    const float* __restrict__ gre,
    const float* __restrict__ gim,
    float* __restrict__ out)
{
    __shared__ float red[128];
    const int i = threadIdx.x;
    float tot = 0.f, sig = 0.f;
    for (int j = 0; j < 128; ++j) {
        const float re = gre[i * 128 + j];
        const float im = gim[i * 128 + j];
        const float mg = sqrtf(re * re + im * im);
        tot += mg;
        if (j == i) sig = mg;
    }
    const float interf = tot - sig;
    red[i] = sig / (interf + N0F);
    __syncthreads();
    for (int s = 64; s > 0; s >>= 1) {
        if (i < s) red[i] += red[i + s];
        __syncthreads();
    }
    if (i == 0) out[0] = -red[0] * 1e6f;
}

// ---------------------------------------------------------------------------
extern "C" void kernel_launch(void* const* d_in, const int* in_sizes, int n_in,
                              void* d_out, int out_size, void* d_ws, size_t ws_size,
                              hipStream_t stream)
{
    const float* Wv   = (const float*)d_in[0];
    const float* Are  = (const float*)d_in[1];
    const float* Aim  = (const float*)d_in[2];
    const float* hdre = (const float*)d_in[3];
    const float* hdim = (const float*)d_in[4];

    float* ws  = (float*)d_ws;
    float* hre = ws + WS_HRE;
    float* him = ws + WS_HIM;
    float* gre = ws + WS_GRE;
    float* gim = ws + WS_GIM;

    k1_hvec<<<NROWS / (16 * 8), 256, 0, stream>>>(Wv, Are, Aim, hdre, hdim, hre, him);
    k2_gemm<<<16, 128, 0, stream>>>(Wv, hre, him, gre, gim);
    k3_rate<<<1, 128, 0, stream>>>(gre, gim, (float*)d_out);
}